// LogicConv3d_4440996184573
// MI455X (gfx1250) — compile-verified
//
#include <hip/hip_runtime.h>
#include <stdint.h>

// Problem constants (match the reference)
#define B_   4
#define C_   3
#define H_   32
#define K_   16
#define S_   16
#define P_   27000            // 30*30*30
#define HWD  (H_*H_*H_)       // 32768
#define CHWD (C_*HWD)         // 98304  (per-batch x stride)
#define TP   128              // positions (=threads) per workgroup

typedef __attribute__((ext_vector_type(2))) float v2f;
typedef __attribute__((ext_vector_type(8))) float v8f;

// ---------------------------------------------------------------------------
// Kernel 1: coef[row][c] = sum_o softmax(w_row)[o] * M_OPS[o][c]
// 496 rows (levels concatenated: 256+128+64+32+16 = n*K entries).
// One wave per 16-row chunk; the K=16 contraction runs as 4 chained
// V_WMMA_F32_16X16X4_F32 ops (A = softmax probs 16x4 chunk, B = M_OPS 4x16,
// columns 4..15 zero). M_OPS[r][c] closed form over row bits b0..b3:
//   c0 = b3, c1 = b1-b3, c2 = b2-b3, c3 = b0-b1-b2+b3
// With j unrolled only `half` (lane>>4) is runtime -> pure v_cndmask, no
// dynamic register indexing, EXEC all-ones into the WMMAs.
// ---------------------------------------------------------------------------
__global__ void __launch_bounds__(32)
coef_wmma_kernel(const float* __restrict__ w0, const float* __restrict__ w1,
                 const float* __restrict__ w2, const float* __restrict__ w3,
                 const float* __restrict__ w4, float* __restrict__ coef)
{
  const int lane = threadIdx.x;       // 0..31
  const int half = lane >> 4;         // K-half per ISA 32-bit A/B layouts
  const int rloc = lane & 15;         // row within chunk / N column
  const int chunk = blockIdx.x;       // 0..30 (level boundaries are 16-aligned)
  const int row = chunk * 16 + rloc;  // global row 0..495

  const float* wp; int local;         // uniform per block (level uniform per chunk)
  if (row < 256)      { wp = w0; local = row;       }
  else if (row < 384) { wp = w1; local = row - 256; }
  else if (row < 448) { wp = w2; local = row - 384; }
  else if (row < 480) { wp = w3; local = row - 448; }
  else                { wp = w4; local = row - 480; }

  // softmax over the 16 weights of this row (all static indexing)
  float wr[16];
  float m = -1e30f;
  #pragma unroll
  for (int o = 0; o < 16; ++o) { wr[o] = wp[local * 16 + o]; m = fmaxf(m, wr[o]); }
  float z = 0.f;
  #pragma unroll
  for (int o = 0; o < 16; ++o) { wr[o] = __expf(wr[o] - m); z += wr[o]; }
  const float inv = 1.f / z;
  #pragma unroll
  for (int o = 0; o < 16; ++o) wr[o] *= inv;

  v8f acc = {};
  const int   col = rloc;
  const float fh  = (float)half;
  #pragma unroll
  for (int j = 0; j < 4; ++j) {
    // A element rows: r = 4j + 2*half (VGPR0 / .x), r+1 (VGPR1 / .y)
    v2f A, Bv;
    A.x = half ? wr[4*j + 2] : wr[4*j + 0];
    A.y = half ? wr[4*j + 3] : wr[4*j + 1];
    // M_OPS closed form; row bits: b0 = 0(.x)/1(.y), b1 = half, b2 = j&1, b3 = j>>1
    const float b2  = (float)(j & 1);
    const float b3  = (float)(j >> 1);
    const float c1v = fh - b3;
    const float c2v = b2 - b3;
    const float c3x = b3 - fh - b2;      // b0 = 0
    const float c3y = c3x + 1.f;         // b0 = 1
    Bv.x = (col == 0) ? b3 : (col == 1) ? c1v : (col == 2) ? c2v : (col == 3) ? c3x : 0.f;
    Bv.y = (col == 0) ? b3 : (col == 1) ? c1v : (col == 2) ? c2v : (col == 3) ? c3y : 0.f;
    acc = __builtin_amdgcn_wmma_f32_16x16x4_f32(false, A, false, Bv,
                                                (short)0, acc, false, false);
  }

  // D layout: VGPR i -> M = half*8 + i, N = lane&15. Only N<4 is meaningful.
  if (col < 4) {
    #pragma unroll
    for (int i = 0; i < 8; ++i) {
      const int gr = chunk * 16 + half * 8 + i;
      coef[gr * 4 + col] = acc[i];
    }
  }
}

// ---------------------------------------------------------------------------
// Kernel 2: main gather + binop tree.
// Workgroup = (k, 128 positions). Index slab (128*16 int4 per array) is
// contiguous in memory -> streamed to LDS with async global->LDS B128 copies
// (non-temporal: read-once 221 MB stream, keep L2 for the hot 1.5 MB x).
// Each thread evaluates one position for all 4 batches (4x index reuse).
// ---------------------------------------------------------------------------
__global__ void __launch_bounds__(TP)
logic_conv_kernel(const float* __restrict__ x,
                  const int*   __restrict__ idxA,
                  const int*   __restrict__ idxB,
                  const float* __restrict__ coef,
                  float*       __restrict__ out)
{
  __shared__ int4   sA[TP * S_];   // 32 KB
  __shared__ int4   sB[TP * S_];   // 32 KB
  __shared__ float4 sLut[31];      // per-k coefficient LUT (levels concatenated)

  const int t  = threadIdx.x;
  const int k  = blockIdx.y;
  const int p0 = blockIdx.x * TP;

  // ---- async stage of index tiles (CDNA5 GLOBAL_LOAD_ASYNC_TO_LDS_B128) ----
  const uint32_t base16 = ((uint32_t)k * (uint32_t)P_ + (uint32_t)p0) * (uint32_t)S_; // int4 units
  const uint32_t max16  = (uint32_t)K_ * (uint32_t)P_ * (uint32_t)S_ - 1u;            // last valid int4
  const uint32_t ldsA = (uint32_t)(uintptr_t)(&sA[0]);
  const uint32_t ldsB = (uint32_t)(uintptr_t)(&sB[0]);
  #pragma unroll 4
  for (int it = 0; it < S_; ++it) {
    const uint32_t c16 = (uint32_t)(it * TP + t);
    uint32_t g16 = base16 + c16;
    if (g16 > max16) g16 = max16;              // clamp: tail lanes stay in-bounds (unused)
    const uint32_t goff = g16 * 16u;
    asm volatile("global_load_async_to_lds_b128 %0, %1, %2 th:TH_LOAD_NT"
                 :: "v"(ldsA + c16 * 16u), "v"(goff), "s"(idxA) : "memory");
    asm volatile("global_load_async_to_lds_b128 %0, %1, %2 th:TH_LOAD_NT"
                 :: "v"(ldsB + c16 * 16u), "v"(goff), "s"(idxB) : "memory");
  }

  // ---- per-k coefficient LUT: 31 entries = levels (16,8,4,2,1) ----
  if (t < 31) {
    int base, n;
    if (t < 16)      { base = 0;   n = t;      }
    else if (t < 24) { base = 256; n = t - 16; }
    else if (t < 28) { base = 384; n = t - 24; }
    else if (t < 30) { base = 448; n = t - 28; }
    else             { base = 480; n = 0;      }
    sLut[t] = ((const float4*)coef)[base + n * K_ + k];
  }

  asm volatile("s_wait_asynccnt 0x0" ::: "memory");  // this wave's async copies done
  __syncthreads();                                   // all waves' copies visible

  const int p = p0 + t;
  if (p < P_) {
    float v[B_][16];
    const int4* ma = &sA[t * S_];
    const int4* mb = &sB[t * S_];

    // level 0: gather + leaf binops (indices: .x=h .y=w .z=d .w=c)
    #pragma unroll
    for (int s = 0; s < S_; ++s) {
      const int4 ia = ma[s];
      const int4 ib = mb[s];
      const int offA = ia.w * HWD + ia.x * 1024 + ia.y * 32 + ia.z;
      const int offB = ib.w * HWD + ib.x * 1024 + ib.y * 32 + ib.z;
      const float4 cc = sLut[s];
      #pragma unroll
      for (int bb = 0; bb < B_; ++bb) {
        const float a = x[bb * CHWD + offA];
        const float b = x[bb * CHWD + offB];
        v[bb][s] = fmaf(cc.w, a * b, fmaf(cc.z, b, fmaf(cc.y, a, cc.x)));
      }
    }

    // levels 1..4: pairwise tree reduction, coefficients from LUT
    int lut = 16;
    #pragma unroll
    for (int n = 8; n >= 1; n >>= 1) {
      #pragma unroll
      for (int i = 0; i < n; ++i) {
        const float4 cc = sLut[lut + i];
        #pragma unroll
        for (int bb = 0; bb < B_; ++bb) {
          const float a = v[bb][2 * i];
          const float b = v[bb][2 * i + 1];
          v[bb][i] = fmaf(cc.w, a * b, fmaf(cc.z, b, fmaf(cc.y, a, cc.x)));
        }
      }
      lut += n;
    }

    // out shape (B, K, P): coalesced along p, write-once -> non-temporal
    #pragma unroll
    for (int bb = 0; bb < B_; ++bb)
      __builtin_nontemporal_store(v[bb][0],
          &out[((size_t)bb * K_ + k) * (size_t)P_ + (size_t)p]);
  }
}

// ---------------------------------------------------------------------------
extern "C" void kernel_launch(void* const* d_in, const int* in_sizes, int n_in,
                              void* d_out, int out_size, void* d_ws, size_t ws_size,
                              hipStream_t stream)
{
  (void)in_sizes; (void)n_in; (void)out_size; (void)ws_size;
  const float* x    = (const float*)d_in[0];
  const int*   idxA = (const int*)d_in[1];
  const int*   idxB = (const int*)d_in[2];
  const float* w0   = (const float*)d_in[3];
  const float* w1   = (const float*)d_in[4];
  const float* w2   = (const float*)d_in[5];
  const float* w3   = (const float*)d_in[6];
  const float* w4   = (const float*)d_in[7];

  float* coef = (float*)d_ws;   // 496 rows x float4 = 7936 B

  coef_wmma_kernel<<<dim3(31), dim3(32), 0, stream>>>(w0, w1, w2, w3, w4, coef);

  dim3 grid((P_ + TP - 1) / TP, K_);
  logic_conv_kernel<<<grid, dim3(TP), 0, stream>>>(x, idxA, idxB, coef, (float*)d_out);
}